// TGSM_455266533610
// MI455X (gfx1250) — compile-verified
//
#include <hip/hip_runtime.h>
#include <math.h>

// Problem constants (match reference)
#define TT 256
#define NN 14
#define FD 4
#define HH 32
#define KK 5
#define CC 2

typedef __attribute__((ext_vector_type(2))) float v2f;
typedef __attribute__((ext_vector_type(8))) float v8f;

static __device__ __forceinline__ v8f wmma_k4(v2f a, v2f b, v8f c) {
  // D = A(16x4) * B(4x16) + C
  return __builtin_amdgcn_wmma_f32_16x16x4_f32(false, a, false, b, (short)0, c, false, false);
}

// fast hardware math (v_rcp_f32 / v_rsq_f32 / v_sqrt_f32)
static __device__ __forceinline__ float frcp(float x) { return __builtin_amdgcn_rcpf(x); }
static __device__ __forceinline__ float frsq(float x) { return __builtin_amdgcn_rsqf(x); }
static __device__ __forceinline__ float fsqrt(float x) { return __builtin_amdgcn_sqrtf(x); }

// One butterfly-add stage via DPP (no LDS traffic)
template <int CTRL>
static __device__ __forceinline__ float dppadd(float x) {
  int v = __builtin_amdgcn_update_dpp(0, __float_as_int(x), CTRL, 0xF, 0xF, true);
  return x + __int_as_float(v);
}
// sum over the 16 lanes of this half-wave group (commutative-reduction DPP ladder)
static __device__ __forceinline__ float xorsum16(float x) {
  x = dppadd<0xB1>(x);   // quad_perm(1,0,3,2)  ~ xor 1
  x = dppadd<0x4E>(x);   // quad_perm(2,3,0,1)  ~ xor 2
  x = dppadd<0x141>(x);  // row_half_mirror     ~ cross-quad in 8
  x = dppadd<0x140>(x);  // row_mirror          ~ cross-8 in 16
  return x;
}

__global__ __launch_bounds__(32)
void tgsm_kernel(const float* __restrict__ win,
                 const float* __restrict__ gate_w, const float* __restrict__ gate_b,
                 const float* __restrict__ gcn_w, const float* __restrict__ gcn_b,
                 const float* __restrict__ ln_g,  const float* __restrict__ ln_b,
                 const float* __restrict__ attn_w, const float* __restrict__ attn_b,
                 const float* __restrict__ c1w, const float* __restrict__ c1b,
                 const float* __restrict__ c2w, const float* __restrict__ c2b,
                 const float* __restrict__ c3w, const float* __restrict__ c3b,
                 float* __restrict__ out) {
  __shared__ __align__(16) float lm[16 * 16];        // row-scaled a-matrix / esm scratch
  __shared__ __align__(16) float ls[16 * HH];        // scaled-sup scratch
  __shared__ __align__(16) float lc[KK + HH + 64 + 32 + 3]; // combined + h1 + h2

  const int b   = blockIdx.x;
  const int tid = threadIdx.x;
  const int col = tid & 15;   // N (or K/H) index within half-wave
  const int hi  = tid >> 4;   // half-wave group

  // ---- preload small constant weights (uniform or per-lane) ----
  const float gw0 = gate_w[0], gw1 = gate_w[1], gbias = gate_b[0];
  v2f gB0, gB1;  // gcn_w (4x32) in B-operand layout, tiles H=[0..15],[16..31]
  gB0.x = gcn_w[(2 * hi) * HH + col];      gB0.y = gcn_w[(2 * hi + 1) * HH + col];
  gB1.x = gcn_w[(2 * hi) * HH + col + 16]; gB1.y = gcn_w[(2 * hi + 1) * HH + col + 16];
  const float gb0 = gcn_b[col], gb1 = gcn_b[col + 16];
  const float lg0 = ln_g[col],  lg1 = ln_g[col + 16];
  const float lb0 = ln_b[col],  lb1 = ln_b[col + 16];
  const float aw0 = attn_w[col], aw1 = attn_w[col + 16];
  const float ab  = attn_b[0];

  // edge-state matrix in WMMA C-layout: reg j, lane l -> esm[j+8*hi][col]
  v8f esm = {0, 0, 0, 0, 0, 0, 0, 0};
  // online softmax state for temporal attention
  float M = -__builtin_inff(), S = 0.f, PA = 0.f, PB = 0.f;

  const float* base = win + (size_t)b * TT * NN * FD;
  const int lofs = col * FD + 2 * hi;   // per-lane offset within a (t) slab

  // software pipeline: preload step 0's de slab
  v2f de_next = {0.f, 0.f};
  if (col < NN) de_next = *(const v2f*)(base + lofs);

#pragma unroll 1
  for (int t = 0; t < TT; ++t) {
    v2f de = de_next;
    // stream ahead: prefetch ~8 slabs (~1.8KB) down the line
    __builtin_prefetch(base + (t + 8) * (NN * FD), 0, 0);
    // issue next step's load before the serial math chain of this step
    {
      int tn = (t + 1 < TT) ? (t + 1) : (TT - 1);
      if (col < NN) de_next = *(const v2f*)(base + tn * (NN * FD) + lofs);
    }

    // row L2 norms (4 comps split across lane pair l <-> l^16)
    float pr   = de.x * de.x + de.y * de.y;
    float full = pr + __shfl_xor(pr, 16, 32);
    float inv  = frsq(fmaxf(full, 1e-24f));   // == 1/max(sqrt(full),1e-12)
    v2f nrm; nrm.x = de.x * inv; nrm.y = de.y * inv;

    v8f zero8 = {0, 0, 0, 0, 0, 0, 0, 0};
    // cosine similarity: nrm (16x4) @ nrm^T (4x16) — A and B are the SAME lane regs
    v8f craw = wmma_k4(nrm, nrm, zero8);

    v8f  areg;
    float dis[8];
#pragma unroll
    for (int j = 0; j < 8; ++j) {
      int  m     = j + 8 * hi;
      bool valid = (m < NN) && (col < NN) && (m != col);
      float adj  = valid ? (craw[j] + 1.f) * 0.5f : 0.f;
      float e    = esm[j];
      float z    = frcp(1.f + __expf(-(gw0 * e + gw1 * adj + gbias)));  // sigmoid
      e          = e + z * (adj - e);
      esm[j]     = e;
      float a    = e + ((m == col && m < NN) ? 1.f : 0.f);   // + eye (14x14 only)
      areg[j]    = a;
      float d    = xorsum16(a);                              // row degree (DPP ladder)
      dis[j]     = frsq(fmaxf(d, 1e-6f));                    // D^{-1/2}
    }

    // sup = de @ gcn_w + b
    v8f s0 = wmma_k4(de, gB0, zero8);
    v8f s1 = wmma_k4(de, gB1, zero8);
#pragma unroll
    for (int j = 0; j < 8; ++j) {
      int m = j + 8 * hi;
      ls[m * HH + col]      = (s0[j] + gb0) * dis[j];        // D^{-1/2} * sup   (rows = n)
      ls[m * HH + col + 16] = (s1[j] + gb1) * dis[j];
      lm[m * 16 + col]      = areg[j] * dis[j];              // D^{-1/2} * a     (rows = m)
    }
    __syncthreads();

    // h = (D^-1/2 a)(16x16) @ (D^-1/2 sup)(16x32): 4 K-chunks x 2 H-tiles
    v8f h0 = zero8, h1 = zero8;
#pragma unroll
    for (int c = 0; c < 4; ++c) {
      v2f aop = *(const v2f*)&lm[col * 16 + 4 * c + 2 * hi]; // A chunk (transposed read)
      int kr  = 4 * c + 2 * hi;
      v2f b0; b0.x = ls[kr * HH + col];      b0.y = ls[(kr + 1) * HH + col];
      v2f b1; b1.x = ls[kr * HH + col + 16]; b1.y = ls[(kr + 1) * HH + col + 16];
      h0 = wmma_k4(aop, b0, h0);
      h1 = wmma_k4(aop, b1, h1);
    }
    __syncthreads();

    // LayerNorm over H=32 ; ELU ; node-mean accumulation
    float ge0p = 0.f, ge1p = 0.f;
#pragma unroll
    for (int j = 0; j < 8; ++j) {
      int   m  = j + 8 * hi;
      float x0 = h0[j], x1 = h1[j];
      float mu = xorsum16(x0 + x1) * (1.f / 32.f);
      float s2 = xorsum16(x0 * x0 + x1 * x1) * (1.f / 32.f);
      float var  = fmaxf(s2 - mu * mu, 0.f);
      float rinv = frsq(var + 1e-5f);
      x0 = (x0 - mu) * rinv * lg0 + lb0;
      x1 = (x1 - mu) * rinv * lg1 + lb1;
      x0 = (x0 > 0.f) ? x0 : (__expf(x0) - 1.f);
      x1 = (x1 > 0.f) ? x1 : (__expf(x1) - 1.f);
      if (m < NN) { ge0p += x0; ge1p += x1; }
    }
    float ge0 = (ge0p + __shfl_xor(ge0p, 16, 32)) * (1.f / 14.f);
    float ge1 = (ge1p + __shfl_xor(ge1p, 16, 32)) * (1.f / 14.f);

    // temporal attention logit + online softmax update (no embs materialization)
    float logit = xorsum16(ge0 * aw0 + ge1 * aw1) + ab;
    float newM  = fmaxf(M, logit);
    float so    = __expf(M - newM);
    float sn    = __expf(logit - newM);
    S  = S * so + sn;
    PA = PA * so + sn * ge0;
    PB = PB * so + sn * ge1;
    M  = newM;
  }

  // ---- pooled features -> combined[5..36] ----
  {
    float rS = frcp(S);
    if (hi == 0) { lc[KK + col] = PA * rS; lc[KK + 16 + col] = PB * rS; }
  }

  // ---- symmetrize final esm into lm ----
#pragma unroll
  for (int j = 0; j < 8; ++j) lm[(j + 8 * hi) * 16 + col] = esm[j];
  __syncthreads();
  float symv[8];
#pragma unroll
  for (int j = 0; j < 8; ++j) {
    int m = j + 8 * hi;
    symv[j] = 0.5f * (lm[m * 16 + col] + lm[col * 16 + m]);
  }
  __syncthreads();
#pragma unroll
  for (int j = 0; j < 8; ++j) lm[(j + 8 * hi) * 16 + col] = symv[j];
  __syncthreads();

  // ---- cyclic Jacobi eigenvalues of the 14x14 symmetric matrix ----
#pragma unroll 1
  for (int sweep = 0; sweep < 10; ++sweep) {
#pragma unroll 1
    for (int p = 0; p < NN - 1; ++p) {
#pragma unroll 1
      for (int q = p + 1; q < NN; ++q) {
        float app = lm[p * 16 + p], aqq = lm[q * 16 + q], apq = lm[p * 16 + q];
        __syncthreads();
        if (fabsf(apq) > 1e-12f) {   // uniform: all lanes read same LDS value
          float tau = (aqq - app) * frcp(2.f * apq);
          float rt  = fsqrt(1.f + tau * tau);
          float tr  = (tau >= 0.f) ? frcp(tau + rt) : frcp(tau - rt);
          float cj  = frsq(1.f + tr * tr);
          float sj  = tr * cj;
          if (tid < NN) {            // column rotation A <- A*J
            float vp = lm[tid * 16 + p], vq = lm[tid * 16 + q];
            lm[tid * 16 + p] = cj * vp - sj * vq;
            lm[tid * 16 + q] = sj * vp + cj * vq;
          }
          __syncthreads();
          if (tid < NN) {            // row rotation A <- J^T*A
            float vp = lm[p * 16 + tid], vq = lm[q * 16 + tid];
            lm[p * 16 + tid] = cj * vp - sj * vq;
            lm[q * 16 + tid] = sj * vp + cj * vq;
          }
          __syncthreads();
        }
      }
    }
  }

  // ---- top-K eigenvalues, ascending -> combined[0..4] ----
  float dv = (col < NN) ? lm[col * 16 + col] : -__builtin_inff();
#pragma unroll
  for (int i = 0; i < KK; ++i) {
    float mx = dv;
    mx = fmaxf(mx, __shfl_xor(mx, 1, 32));
    mx = fmaxf(mx, __shfl_xor(mx, 2, 32));
    mx = fmaxf(mx, __shfl_xor(mx, 4, 32));
    mx = fmaxf(mx, __shfl_xor(mx, 8, 32));
    if (tid == 0) lc[KK - 1 - i] = mx;
    int cand = (dv == mx) ? col : 31;                 // knock out lowest matching col
    cand = min(cand, __shfl_xor(cand, 1, 32));
    cand = min(cand, __shfl_xor(cand, 2, 32));
    cand = min(cand, __shfl_xor(cand, 4, 32));
    cand = min(cand, __shfl_xor(cand, 8, 32));
    if (col == cand) dv = -__builtin_inff();
  }
  __syncthreads();

  // ---- classifier MLP: 37 -> 64 -> 32 -> 2 ----
  {
    float a0 = c1b[tid], a1 = c1b[tid + 32];
    for (int i = 0; i < KK + HH; ++i) {
      float x = lc[i];
      a0 += x * c1w[i * 64 + tid];
      a1 += x * c1w[i * 64 + tid + 32];
    }
    a0 = (a0 > 0.f) ? a0 : (__expf(a0) - 1.f);
    a1 = (a1 > 0.f) ? a1 : (__expf(a1) - 1.f);
    lc[37 + tid] = a0;
    lc[37 + tid + 32] = a1;
  }
  __syncthreads();
  {
    float a = c2b[tid];
    for (int i = 0; i < 64; ++i) a += lc[37 + i] * c2w[i * 32 + tid];
    a = (a > 0.f) ? a : (__expf(a) - 1.f);
    lc[37 + 64 + tid] = a;
  }
  __syncthreads();
  if (tid < CC) {
    float a = c3b[tid];
    for (int i = 0; i < 32; ++i) a += lc[37 + 64 + i] * c3w[i * CC + tid];
    out[(size_t)b * CC + tid] = a;
  }
}

extern "C" void kernel_launch(void* const* d_in, const int* in_sizes, int n_in,
                              void* d_out, int out_size, void* d_ws, size_t ws_size,
                              hipStream_t stream) {
  (void)n_in; (void)d_ws; (void)ws_size; (void)out_size;
  const float* win    = (const float*)d_in[0];
  const float* gate_w = (const float*)d_in[1];
  const float* gate_b = (const float*)d_in[2];
  const float* gcn_w  = (const float*)d_in[3];
  const float* gcn_b  = (const float*)d_in[4];
  const float* ln_g   = (const float*)d_in[5];
  const float* ln_b   = (const float*)d_in[6];
  const float* attn_w = (const float*)d_in[7];
  const float* attn_b = (const float*)d_in[8];
  const float* c1w    = (const float*)d_in[9];
  const float* c1b    = (const float*)d_in[10];
  const float* c2w    = (const float*)d_in[11];
  const float* c2b    = (const float*)d_in[12];
  const float* c3w    = (const float*)d_in[13];
  const float* c3b    = (const float*)d_in[14];
  float* out = (float*)d_out;

  int B = in_sizes[0] / (TT * NN * FD);  // 1024

  tgsm_kernel<<<dim3(B), dim3(32), 0, stream>>>(
      win, gate_w, gate_b, gcn_w, gcn_b, ln_g, ln_b, attn_w, attn_b,
      c1w, c1b, c2w, c2b, c3w, c3b, out);
}